// PeerLookupQueryUnit_55473797595869
// MI455X (gfx1250) — compile-verified
//
#include <hip/hip_runtime.h>

// MI455X / gfx1250, wave32. fp32 GEMM via V_WMMA_F32_16X16X4_F32 + streaming top-16.

typedef __attribute__((ext_vector_type(2))) float v2f;
typedef __attribute__((ext_vector_type(8))) float v8f;

#define NUM_EMBED 65536
#define EMB_DIM   256
#define TOPK      16
#define TOKENS    4096
#define WAVES     8
#define KSTEPS    (EMB_DIM / 4)
#define NEG_INF   (-3.402823466e+38f)

// Bitonic sort of 16 values (one per lane within each 16-lane half), descending.
__device__ __forceinline__ void sort16_desc(float& v, int& i, int lid) {
#pragma unroll
  for (int k = 2; k <= 16; k <<= 1) {
#pragma unroll
    for (int j = k >> 1; j > 0; j >>= 1) {
      float pv = __shfl_xor(v, j, 32);
      int   pi = __shfl_xor(i, j, 32);
      bool up    = ((lid & k) == 0);
      bool lower = ((lid & j) == 0);
      bool takeMax = (up == lower);
      bool sw = takeMax ? (pv > v) : (pv < v);
      v = sw ? pv : v;
      i = sw ? pi : i;
    }
  }
}

// Clean a bitonic 16-sequence into descending order (4 stages).
__device__ __forceinline__ void bitonic_clean16_desc(float& v, int& i, int lid) {
#pragma unroll
  for (int j = 8; j > 0; j >>= 1) {
    float pv = __shfl_xor(v, j, 32);
    int   pi = __shfl_xor(i, j, 32);
    bool lower = ((lid & j) == 0);
    bool sw = lower ? (pv > v) : (pv < v);
    v = sw ? pv : v;
    i = sw ? pi : i;
  }
}

__global__ __launch_bounds__(256, 1)
void PeerLookupQueryUnit_topk_kernel(const float* __restrict__ x,
                                     const float* __restrict__ W,
                                     float* __restrict__ outv,
                                     int* __restrict__ outi) {
  __shared__ float s_val[WAVES][16][TOPK];
  __shared__ int   s_idx[WAVES][16][TOPK];

  const int lane  = threadIdx.x & 31;
  const int wave  = threadIdx.x >> 5;
  const int lid   = lane & 15;      // position within 16-lane half
  const int khalf = lane >> 4;      // which half-wave (selects K pair / row group)
  const int t0    = blockIdx.x * 16;

  // ---- Preload all A fragments (16 tokens x 256 dims) into registers.
  // A 16x4 f32 layout: lanes 0-15 hold K={k0,k0+1}, lanes 16-31 hold K={k0+2,k0+3}.
  v2f af[KSTEPS];
  const float* xr = x + (size_t)(t0 + lid) * EMB_DIM + 2 * khalf;
#pragma unroll
  for (int ks = 0; ks < KSTEPS; ++ks)
    af[ks] = *(const v2f*)(xr + ks * 4);

  // Running per-row top-16, sorted descending across the 16 lanes of each half.
  float topv[8]; int topi[8];
#pragma unroll
  for (int r = 0; r < 8; ++r) { topv[r] = NEG_INF; topi[r] = 0; }

  // B 4x16 layout mirrors A: lane loads W[n0 + lid][4*ks + 2*khalf .. +1] as b64.
  const float* wlane = W + (size_t)lid * EMB_DIM + 2 * khalf;

  for (int tile = wave; tile < NUM_EMBED / 16; tile += WAVES) {
    const int n0 = tile << 4;
    const float* wb = wlane + (size_t)n0 * EMB_DIM;
    // Prefetch this lane's row for the wave's next tile (gfx1250 global_prefetch_b8).
    __builtin_prefetch(wb + (size_t)(16 * WAVES) * EMB_DIM, 0, 3);

    v8f c = {};
#pragma unroll
    for (int ks = 0; ks < KSTEPS; ++ks) {
      v2f b = *(const v2f*)(wb + ks * 4);
      c = __builtin_amdgcn_wmma_f32_16x16x4_f32(
          /*neg_a=*/false, af[ks], /*neg_b=*/false, b,
          /*c_mod=*/(short)0, c, /*reuse_a=*/false, /*reuse_b=*/false);
    }

    // Merge each 16-score row into its running top-16 (threshold-gated, scalar branch
    // via ballot so EXEC stays all-ones around the WMMA loop).
#pragma unroll
    for (int r = 0; r < 8; ++r) {
      float nv = c[r];
      int   ni = n0 + lid;
      float thr = __shfl(topv[r], 15, 16);  // current 16th-largest, per half
      if (__builtin_amdgcn_ballot_w32(nv > thr)) {
        sort16_desc(nv, ni, lid);
        // reverse new list (lane l <-> l^15), pairwise max -> bitonic top-16 of union
        float rv = __shfl_xor(nv, 15, 32);
        int   ri = __shfl_xor(ni, 15, 32);
        bool take = rv > topv[r];
        float mv = take ? rv : topv[r];
        int   mi = take ? ri : topi[r];
        bitonic_clean16_desc(mv, mi, lid);
        topv[r] = mv; topi[r] = mi;
      }
    }
  }

  // ---- Publish per-wave sorted top-16 lists to LDS (8 waves x 16 rows x 16 slots).
#pragma unroll
  for (int r = 0; r < 8; ++r) {
    const int row = r + 8 * khalf;
    s_val[wave][row][lid] = topv[r];
    s_idx[wave][row][lid] = topi[r];
  }
  __syncthreads();

  // ---- Final reduce: per token row, 128 candidates -> top-16 (wave per 2 rows).
#pragma unroll
  for (int rr = 0; rr < 2; ++rr) {
    const int row = wave * 2 + rr;
    const int tok = t0 + row;
    float v[4]; int id[4];
#pragma unroll
    for (int q = 0; q < 4; ++q) {
      const int cnd = lane + 32 * q;      // 0..127
      v[q]  = s_val[cnd >> 4][row][cnd & 15];
      id[q] = s_idx[cnd >> 4][row][cnd & 15];
    }
    for (int it = 0; it < TOPK; ++it) {
      float bv = v[0]; int bi = id[0];
#pragma unroll
      for (int q = 1; q < 4; ++q) {
        bool t = (v[q] > bv) || (v[q] == bv && id[q] < bi);
        bv = t ? v[q] : bv; bi = t ? id[q] : bi;
      }
#pragma unroll
      for (int off = 16; off > 0; off >>= 1) {
        float pv = __shfl_xor(bv, off, 32);
        int   pi = __shfl_xor(bi, off, 32);
        bool t = (pv > bv) || (pv == bv && pi < bi);
        bv = t ? pv : bv; bi = t ? pi : bi;
      }
      if (lane == 0) {
        outv[tok * TOPK + it] = bv;
        outi[tok * TOPK + it] = bi;
      }
#pragma unroll
      for (int q = 0; q < 4; ++q)
        if (v[q] == bv && id[q] == bi) v[q] = NEG_INF;
    }
  }
}

extern "C" void kernel_launch(void* const* d_in, const int* in_sizes, int n_in,
                              void* d_out, int out_size, void* d_ws, size_t ws_size,
                              hipStream_t stream) {
  const float* x = (const float*)d_in[0];   // [2,2048,256] fp32
  const float* W = (const float*)d_in[1];   // [65536,256] fp32
  float* outv = (float*)d_out;                       // values [4096,16] fp32
  int*   outi = (int*)d_out + (size_t)TOKENS * TOPK; // indices [4096,16] int32 (bit-stored)

  dim3 grid(TOKENS / 16);
  dim3 block(256);
  PeerLookupQueryUnit_topk_kernel<<<grid, block, 0, stream>>>(x, W, outv, outi);
}